// RGCN_43619687858916
// MI455X (gfx1250) — compile-verified
//
#include <hip/hip_runtime.h>
#include <hip/hip_bf16.h>

typedef __attribute__((ext_vector_type(2))) float v2f;
typedef __attribute__((ext_vector_type(8))) float v8f;

// ---------------------------------------------------------------- utilities

__global__ void zero_f32(float* __restrict__ p, long long n) {
    long long i = (long long)blockIdx.x * blockDim.x + threadIdx.x;
    if (i < n) p[i] = 0.0f;
}

// deg accumulation: rs_out[r*N + src], rs_in[r*N + dst] += 1
__global__ void degrees_kernel(const int* __restrict__ src, const int* __restrict__ dst,
                               float* __restrict__ rs_out, float* __restrict__ rs_in,
                               int E, int N, long long total) {
    long long i = (long long)blockIdx.x * blockDim.x + threadIdx.x;
    if (i >= total) return;
    int r = (int)(i / E);
    int s = src[i];
    int d = dst[i];
    unsafeAtomicAdd(rs_out + (size_t)r * N + s, 1.0f);
    unsafeAtomicAdd(rs_in  + (size_t)r * N + d, 1.0f);
}

// in-place deg -> rsqrt(max(deg,1))
__global__ void finalize_rs(float* __restrict__ p, long long n) {
    long long i = (long long)blockIdx.x * blockDim.x + threadIdx.x;
    if (i < n) p[i] = rsqrtf(fmaxf(p[i], 1.0f));
}

// ---------------------------------------------------------------- WMMA GEMM
// T[row, col] = sum_k ( X[row,k] * rs[row] ) * W[k, col]
// One wave (32 lanes) computes one 16x16 tile of T via V_WMMA_F32_16X16X4_F32.
__global__ void __launch_bounds__(128)
gemm_scaled_wmma(const float* __restrict__ X, const float* __restrict__ rs,
                 const float* __restrict__ W, float* __restrict__ T,
                 int nRows, int Fin, int Fout) {
    const int lane = threadIdx.x;      // 0..31 (wave32)
    const int m    = lane & 15;
    const int half = lane >> 4;        // 0: K-pair {0,1}, 1: K-pair {2,3}

    const int colTiles = Fout >> 4;
    const int rowTiles = (nRows + 15) >> 4;
    const int tile = blockIdx.x * blockDim.y + threadIdx.y;   // uniform per wave
    if (tile >= rowTiles * colTiles) return;
    const int rowTile = tile / colTiles;
    const int colTile = tile - rowTile * colTiles;

    int row = rowTile * 16 + m;
    const int rowC = (row < nRows) ? row : (nRows - 1);       // clamp: keep EXEC full
    const int col  = colTile * 16 + m;

    const float scale = rs[rowC];
    const float* __restrict__ xrow = X + (size_t)rowC * Fin;

    v8f c = {};
    for (int k0 = 0; k0 < Fin; k0 += 4) {
        // A 16x4 layout: lane m (half h) holds A[m][2h], A[m][2h+1]
        v2f a, b;
        a[0] = xrow[k0 + 2 * half + 0] * scale;
        a[1] = xrow[k0 + 2 * half + 1] * scale;
        // B 4x16 layout (mirror of A): lane m holds B[2h][m], B[2h+1][m]
        b[0] = W[(size_t)(k0 + 2 * half + 0) * Fout + col];
        b[1] = W[(size_t)(k0 + 2 * half + 1) * Fout + col];
        c = __builtin_amdgcn_wmma_f32_16x16x4_f32(
                /*neg_a=*/false, a, /*neg_b=*/false, b,
                /*c_mod=*/(short)0, c, /*reuse_a=*/false, /*reuse_b=*/false);
    }

    // D layout: VGPR v holds row (v + 8*half), col = lane%16
    #pragma unroll
    for (int v = 0; v < 8; ++v) {
        int rr = rowTile * 16 + v + 8 * half;
        if (rr < nRows) T[(size_t)rr * Fout + col] = c[v];
    }
}

// ---------------------------------------------------------------- scatter
// out[dst[e], :] += T[src[e], :] * rs_in[dst[e]]
// one thread per (edge, float4-quad); F multiple of 4
__global__ void scatter_edges(const float* __restrict__ T, const int* __restrict__ src,
                              const int* __restrict__ dst, const float* __restrict__ rs_in_r,
                              float* __restrict__ out, int E, int F) {
    const int qpe = F >> 2;
    long long idx = (long long)blockIdx.x * blockDim.x + threadIdx.x;
    if (idx >= (long long)E * qpe) return;
    int e = (int)(idx / qpe);
    int q = (int)(idx - (long long)e * qpe);
    int s = src[e];
    int d = dst[e];
    float sc = rs_in_r[d];
    const float4 mv = *(const float4*)(T + (size_t)s * F + q * 4);
    float* o = out + (size_t)d * F + q * 4;
    unsafeAtomicAdd(o + 0, mv.x * sc);
    unsafeAtomicAdd(o + 1, mv.y * sc);
    unsafeAtomicAdd(o + 2, mv.z * sc);
    unsafeAtomicAdd(o + 3, mv.w * sc);
}

// ---------------------------------------------------------------- bias (+relu)
__global__ void bias_relu(float* __restrict__ h, const float* __restrict__ b,
                          int R, int F, long long total) {
    long long i = (long long)blockIdx.x * blockDim.x + threadIdx.x;
    if (i >= total) return;
    int f = (int)(i % F);
    float bb = 0.0f;
    for (int r = 0; r < R; ++r) bb += b[(size_t)r * F + f];
    h[i] = fmaxf(h[i] + bb, 0.0f);
}

__global__ void bias_add(float* __restrict__ h, const float* __restrict__ b,
                         int R, int F, long long total) {
    long long i = (long long)blockIdx.x * blockDim.x + threadIdx.x;
    if (i >= total) return;
    int f = (int)(i % F);
    float bb = 0.0f;
    for (int r = 0; r < R; ++r) bb += b[(size_t)r * F + f];
    h[i] += bb;
}

// ---------------------------------------------------------------- launch

static inline int cdiv(long long a, long long b) { return (int)((a + b - 1) / b); }

extern "C" void kernel_launch(void* const* d_in, const int* in_sizes, int n_in,
                              void* d_out, int out_size, void* d_ws, size_t ws_size,
                              hipStream_t stream) {
    const float* feat = (const float*)d_in[0];   // [N,128]
    const float* W1   = (const float*)d_in[1];   // [3,128,128]
    const float* b1   = (const float*)d_in[2];   // [3,128]
    const float* W2   = (const float*)d_in[3];   // [3,128,16]
    const float* b2   = (const float*)d_in[4];   // [3,16]
    const int*   src  = (const int*)d_in[5];     // [3,E]
    const int*   dst  = (const int*)d_in[6];     // [3,E]
    float* out = (float*)d_out;                  // [N,16]

    const int IN_FEAT = 128, HID = 128, NC = 16, R = 3;
    const int N = in_sizes[0] / IN_FEAT;
    const int E = in_sizes[5] / R;

    float* ws     = (float*)d_ws;
    float* rs_out = ws;                                   // [R,N]
    float* rs_in  = rs_out + (size_t)R * N;               // [R,N]
    float* h1     = rs_in  + (size_t)R * N;               // [N,HID]
    float* t      = h1     + (size_t)N * HID;             // [N,HID] (reused as [N,NC])

    const int TB = 256;

    // --- zero accumulators (ws/out are poisoned by harness)
    zero_f32<<<cdiv(2LL * R * N, TB), TB, 0, stream>>>(rs_out, 2LL * R * N);
    zero_f32<<<cdiv((long long)N * HID, TB), TB, 0, stream>>>(h1, (long long)N * HID);
    zero_f32<<<cdiv((long long)N * NC, TB), TB, 0, stream>>>(out, (long long)N * NC);

    // --- degrees -> rsqrt(max(deg,1)), for all R relations at once
    degrees_kernel<<<cdiv((long long)R * E, TB), TB, 0, stream>>>(
        src, dst, rs_out, rs_in, E, N, (long long)R * E);
    finalize_rs<<<cdiv(2LL * R * N, TB), TB, 0, stream>>>(rs_out, 2LL * R * N);

    const int rowTiles = (N + 15) / 16;
    dim3 gblk(32, 4);   // 4 waves / block, 1 output tile / wave

    // --- layer 1: h1 = sum_r scatter( (feat*rs_out[r]) @ W1[r] ) * rs_in[r][dst]
    for (int r = 0; r < R; ++r) {
        int tiles = rowTiles * (HID / 16);
        gemm_scaled_wmma<<<cdiv(tiles, 4), gblk, 0, stream>>>(
            feat, rs_out + (size_t)r * N, W1 + (size_t)r * IN_FEAT * HID, t,
            N, IN_FEAT, HID);
        scatter_edges<<<cdiv((long long)E * (HID / 4), TB), TB, 0, stream>>>(
            t, src + (size_t)r * E, dst + (size_t)r * E, rs_in + (size_t)r * N,
            h1, E, HID);
    }
    bias_relu<<<cdiv((long long)N * HID, TB), TB, 0, stream>>>(
        h1, b1, R, HID, (long long)N * HID);

    // --- layer 2: out = sum_r scatter( (h1*rs_out[r]) @ W2[r] ) * rs_in[r][dst]
    for (int r = 0; r < R; ++r) {
        int tiles = rowTiles * (NC / 16);
        gemm_scaled_wmma<<<cdiv(tiles, 4), gblk, 0, stream>>>(
            h1, rs_out + (size_t)r * N, W2 + (size_t)r * HID * NC, t,
            N, HID, NC);
        scatter_edges<<<cdiv((long long)E * (NC / 4), TB), TB, 0, stream>>>(
            t, src + (size_t)r * E, dst + (size_t)r * E, rs_in + (size_t)r * N,
            out, E, NC);
    }
    bias_add<<<cdiv((long long)N * NC, TB), TB, 0, stream>>>(
        out, b2, R, NC, (long long)N * NC);
}